// TinyLarimarMemory_4269197492700
// MI455X (gfx1250) — compile-verified
//
#include <hip/hip_runtime.h>
#include <math.h>
#include <stdint.h>

typedef __attribute__((ext_vector_type(2))) float v2f;
typedef __attribute__((ext_vector_type(8))) float v8f;

#define E_DIM 64
#define B_DIM 128
#define C_DIM 384
#define M_DIM 512
#define REG_EPS 1e-6f
#define OBS_STD 0.1f
#define NS_ITERS 14   // Newton-Schulz iterations (even)

#define KC    32      // K-chunk staged per LDS buffer
#define LPITCH 36     // LDS row pitch in floats (144B = multiple of 16B, conflict-free)

// ---------------------------------------------------------------------------
// WMMA wrapper: D = A(16x4 f32) * B(4x16 f32) + C(16x16 f32)
// ---------------------------------------------------------------------------
__device__ __forceinline__ v8f wmma4(v2f a, v2f b, v8f c) {
  return __builtin_amdgcn_wmma_f32_16x16x4_f32(
      /*neg_a=*/false, a, /*neg_b=*/false, b,
      /*c_mod=*/(short)0, c, /*reuse_a=*/false, /*reuse_b=*/false);
}

// ---------------------------------------------------------------------------
// CDNA5 async global->LDS copies (tracked by ASYNCcnt).  Per-lane independent
// global + LDS addresses, so the staging copy can transpose for free.
// ---------------------------------------------------------------------------
__device__ __forceinline__ void async_ld_b128(uint32_t lds_addr, const float* g) {
  asm volatile("global_load_async_to_lds_b128 %0, %1, off"
               :: "v"(lds_addr), "v"(g) : "memory");
}
__device__ __forceinline__ void async_ld_b32(uint32_t lds_addr, const float* g) {
  asm volatile("global_load_async_to_lds_b32 %0, %1, off"
               :: "v"(lds_addr), "v"(g) : "memory");
}
__device__ __forceinline__ void wait_async0() {
  asm volatile("s_wait_asynccnt 0x0" ::: "memory");
}

// ---------------------------------------------------------------------------
// Generic strided-batched GEMM:
//   D[b] = alpha * op(A[b]) @ op(B[b]) + beta * Cin[b] + gamma * I
// Block = 128 threads = 4 waves; block tile 64x64; each wave computes 32x32
// via 2x2 WMMA accumulators.  K-chunks of 32 are double-buffered in LDS via
// async-to-LDS copies.  Both operands staged as S[row][k] (k contiguous):
//  - op moving k to the fast axis of the global layout -> b128 bulk copies
//  - otherwise -> b32 transposing gathers
// All M,N here are multiples of 64; all K multiples of 32.
// ---------------------------------------------------------------------------
template<bool TA, bool TB>
__global__ __launch_bounds__(128) void gemm_wmma_kernel(
    const float* __restrict__ A, long long sAb, int lda,
    const float* __restrict__ B, long long sBb, int ldb,
    const float* __restrict__ Cin, long long sCb, int ldc,
    float* __restrict__ D, long long sDb, int ldd,
    int K, float alpha, float beta, float gamma)
{
  __shared__ float As[2][64][LPITCH];
  __shared__ float Bs[2][64][LPITCH];

  const int tid  = threadIdx.x;
  const int lane = tid & 31;
  const int wave = tid >> 5;
  const int half = lane >> 4;     // 0: lanes 0-15, 1: lanes 16-31
  const int lsub = lane & 15;

  const int bz = blockIdx.z;
  A += (long long)bz * sAb;
  B += (long long)bz * sBb;
  D += (long long)bz * sDb;
  if (Cin) Cin += (long long)bz * sCb;

  const int rowBlk = blockIdx.y * 64;
  const int colBlk = blockIdx.x * 64;

  const uint32_t asBase = (uint32_t)(uintptr_t)(&As[0][0][0]);
  const uint32_t bsBase = (uint32_t)(uintptr_t)(&Bs[0][0][0]);
  const uint32_t bufBytes = 64u * LPITCH * 4u;

  // ---- async stage one K-chunk of A and B into LDS buffer `buf` ----
  auto stage = [&](int chunk, int buf) {
    const int k0c = chunk * KC;
    const uint32_t aL = asBase + (uint32_t)buf * bufBytes;
    const uint32_t bL = bsBase + (uint32_t)buf * bufBytes;
    // A operand: logical A'[m][k], staged As[m][k]
    if (!TA) {
#pragma unroll
      for (int it = 0; it < 4; ++it) {            // 512 x b128 slots
        const int s  = tid + it * 128;
        const int m  = s >> 3;
        const int kq = (s & 7) * 4;
        async_ld_b128(aL + (uint32_t)(m * LPITCH + kq) * 4u,
                      A + (long long)(rowBlk + m) * lda + (k0c + kq));
      }
    } else {
#pragma unroll
      for (int it = 0; it < 16; ++it) {           // 2048 x b32 gathers
        const int s = tid + it * 128;
        const int m = s & 63;
        const int k = s >> 6;
        async_ld_b32(aL + (uint32_t)(m * LPITCH + k) * 4u,
                     A + (long long)(k0c + k) * lda + (rowBlk + m));
      }
    }
    // B operand: logical B'[k][n], staged Bs[n][k]
    if (TB) {
#pragma unroll
      for (int it = 0; it < 4; ++it) {
        const int s  = tid + it * 128;
        const int n  = s >> 3;
        const int kq = (s & 7) * 4;
        async_ld_b128(bL + (uint32_t)(n * LPITCH + kq) * 4u,
                      B + (long long)(colBlk + n) * ldb + (k0c + kq));
      }
    } else {
#pragma unroll
      for (int it = 0; it < 16; ++it) {
        const int s = tid + it * 128;
        const int n = s & 63;
        const int k = s >> 6;
        async_ld_b32(bL + (uint32_t)(n * LPITCH + k) * 4u,
                     B + (long long)(k0c + k) * ldb + (colBlk + n));
      }
    }
  };

  const int rowW = (wave >> 1) * 32;   // block-local wave row origin
  const int colW = (wave & 1) * 32;

  const v8f vzero = {0.f, 0.f, 0.f, 0.f, 0.f, 0.f, 0.f, 0.f};
  v8f acc[2][2];
#pragma unroll
  for (int i = 0; i < 2; ++i)
#pragma unroll
    for (int j = 0; j < 2; ++j) acc[i][j] = vzero;

  const int nChunks = K / KC;
  stage(0, 0);
  for (int t = 0; t < nChunks; ++t) {
    const int cur = t & 1;
    wait_async0();        // this wave's outstanding asyncs (chunk t) done
    __syncthreads();      // all waves done loading chunk t / reading buf cur^1
    if (t + 1 < nChunks) stage(t + 1, cur ^ 1);

    // consume chunk t from LDS
#pragma unroll
    for (int k0 = 0; k0 < KC; k0 += 4) {
      const int ka = k0 + 2 * half;
      v2f afrag[2], bfrag[2];
#pragma unroll
      for (int i = 0; i < 2; ++i)
        afrag[i] = *(const v2f*)&As[cur][rowW + i * 16 + lsub][ka];
#pragma unroll
      for (int j = 0; j < 2; ++j)
        bfrag[j] = *(const v2f*)&Bs[cur][colW + j * 16 + lsub][ka];
#pragma unroll
      for (int i = 0; i < 2; ++i)
#pragma unroll
        for (int j = 0; j < 2; ++j)
          acc[i][j] = wmma4(afrag[i], bfrag[j], acc[i][j]);
    }
  }

  // Epilogue: D layout is VGPR r -> (M = r + 8*half, N = lsub)
#pragma unroll
  for (int i = 0; i < 2; ++i) {
#pragma unroll
    for (int j = 0; j < 2; ++j) {
      const int n = colBlk + colW + j * 16 + lsub;
#pragma unroll
      for (int r = 0; r < 8; ++r) {
        const int m = rowBlk + rowW + i * 16 + r + 8 * half;
        float v = alpha * acc[i][j][r];
        if (Cin) v += beta * Cin[(long long)m * ldc + n];
        if (gamma != 0.0f && m == n) v += gamma;
        D[(long long)m * ldd + n] = v;
      }
    }
  }
}

// ---------------------------------------------------------------------------
// Gauss-Jordan inverse of batched 64x64 SPD matrices, fully in LDS.
// ---------------------------------------------------------------------------
__global__ __launch_bounds__(128) void ginv64_kernel(
    const float* __restrict__ G, float* __restrict__ Ginv)
{
  __shared__ float aug[64][130];
  __shared__ float colk[64];
  const int c = threadIdx.x;       // 0..127
  const int b = blockIdx.x;
  const float* Gb = G + (long long)b * 64 * 64;

  for (int r = 0; r < 64; ++r) {
    float v;
    if (c < 64) v = Gb[r * 64 + c];
    else        v = (c - 64 == r) ? 1.0f : 0.0f;
    aug[r][c] = v;
  }
  __syncthreads();

  for (int k = 0; k < 64; ++k) {
    if (c < 64) colk[c] = aug[c][k];
    __syncthreads();
    const float pinv = 1.0f / colk[k];
    const float rowk = aug[k][c] * pinv;   // column c is private to thread c
    aug[k][c] = rowk;
    for (int r = 0; r < 64; ++r)
      if (r != k) aug[r][c] -= colk[r] * rowk;
    __syncthreads();
  }

  float* Ob = Ginv + (long long)b * 64 * 64;
  for (int r = 0; r < 64; ++r)
    if (c >= 64) Ob[r * 64 + (c - 64)] = aug[r][c];
}

// ---------------------------------------------------------------------------
// Per-batch infinity-norm of an n x n matrix -> scale[b] = 1/||A||_inf^2
// ---------------------------------------------------------------------------
__global__ __launch_bounds__(256) void infnorm_scale_kernel(
    const float* __restrict__ A, long long sAb, int n, float* __restrict__ scale)
{
  __shared__ float red[256];
  const int b = blockIdx.x;
  const float* Ab = A + (long long)b * sAb;
  float mx = 0.0f;
  for (int r = threadIdx.x; r < n; r += 256) {
    float s = 0.0f;
    const float* row = Ab + (long long)r * n;
    for (int cc = 0; cc < n; ++cc) s += fabsf(row[cc]);
    mx = fmaxf(mx, s);
  }
  red[threadIdx.x] = mx;
  __syncthreads();
  for (int off = 128; off > 0; off >>= 1) {
    if (threadIdx.x < off)
      red[threadIdx.x] = fmaxf(red[threadIdx.x], red[threadIdx.x + off]);
    __syncthreads();
  }
  if (threadIdx.x == 0) {
    const float s = red[0];
    scale[b] = 1.0f / (s * s + 1e-30f);
  }
}

__global__ void scale_batched_kernel(const float* __restrict__ A,
                                     float* __restrict__ X,
                                     const float* __restrict__ scale,
                                     long long per_batch, long long total)
{
  long long i = (long long)blockIdx.x * blockDim.x + threadIdx.x;
  if (i >= total) return;
  const int b = (int)(i / per_batch);
  X[i] = A[i] * scale[b];
}

__global__ void axpb_kernel(const float* __restrict__ a,
                            const float* __restrict__ b, float coef,
                            float* __restrict__ out, long long n)
{
  long long i = (long long)blockIdx.x * blockDim.x + threadIdx.x;
  if (i < n) out[i] = a[i] + coef * b[i];
}

__global__ void add_noise_kernel(float* __restrict__ w,
                                 const float* __restrict__ noise,
                                 const float* __restrict__ logvar, long long n)
{
  long long i = (long long)blockIdx.x * blockDim.x + threadIdx.x;
  if (i < n) w[i] += noise[i] * expf(0.5f * logvar[0]);
}

// ---------------------------------------------------------------------------
// Host-side orchestration
// ---------------------------------------------------------------------------
static inline long long cdiv(long long a, long long b) { return (a + b - 1) / b; }

extern "C" void kernel_launch(void* const* d_in, const int* in_sizes, int n_in,
                              void* d_out, int out_size, void* d_ws, size_t ws_size,
                              hipStream_t stream) {
  (void)in_sizes; (void)n_in; (void)out_size; (void)ws_size;
  const float* in_enc   = (const float*)d_in[0];  // [E,B,C]
  const float* q_enc    = (const float*)d_in[1];  // [E,B,C]
  const float* w_noise  = (const float*)d_in[2];  // [E,B,C]
  const float* r_noise  = (const float*)d_in[3];  // [E,B,M]
  const float* mem_mu   = (const float*)d_in[4];  // [M,C]
  const float* w_logvar = (const float*)d_in[5];  // [1]

  const int E = E_DIM, B = B_DIM, C = C_DIM, M = M_DIM;
  const long long EBC = (long long)E * B * C;
  const long long EBM = (long long)E * B * M;

  float* out = (float*)d_out;
  float* retr_out = out;                 // [E,B,C]
  float* wread_out = out + EBC;          // [E,B,M]
  float* newmem_out = out + EBC + EBM;   // [B,M,C]

  float* ws = (float*)d_ws;
  long long off = 0;
  auto alloc = [&](long long n) { float* p = ws + off; off += n; return p; };

  // ---------------- Phase 1: write path ----------------
  float* z      = alloc(EBC);
  float* mmt    = alloc((long long)M * M);
  float* x0s    = alloc((long long)M * M);
  float* x1s    = alloc((long long)M * M);
  float* ys     = alloc((long long)M * M);
  float* projT  = alloc((long long)M * C);
  float* wwr    = alloc(EBM);
  float* G      = alloc((long long)B * 64 * 64);
  float* Ginv   = alloc((long long)B * 64 * 64);
  float* V      = alloc((long long)B * 64 * C);
  float* sc1    = alloc(64);

  // 1) z = input + 0.1*write_noise
  axpb_kernel<<<(unsigned)cdiv(EBC, 256), 256, 0, stream>>>(in_enc, w_noise, OBS_STD, z, EBC);

  // 2) MMT = Mm @ Mm^T + eps*I   (512x512, K=384, NT)
  gemm_wmma_kernel<false, true><<<dim3(M/64, M/64, 1), 128, 0, stream>>>(
      mem_mu, 0, C, mem_mu, 0, C, nullptr, 0, 0, mmt, 0, M, C, 1.0f, 0.0f, REG_EPS);

  // 3-5) MMT_inv via Newton-Schulz.  All iterates are polynomials in the SPD
  // matrix MMT, hence symmetric -> run both GEMMs as NT for b128 staging.
  infnorm_scale_kernel<<<1, 256, 0, stream>>>(mmt, 0, M, sc1);
  scale_batched_kernel<<<(unsigned)cdiv((long long)M*M, 256), 256, 0, stream>>>(
      mmt, x0s, sc1, (long long)M * M, (long long)M * M);
  {
    float* xc = x0s; float* xn = x1s;
    for (int it = 0; it < NS_ITERS; ++it) {
      gemm_wmma_kernel<false, true><<<dim3(M/64, M/64, 1), 128, 0, stream>>>(
          xc, 0, M, mmt, 0, M, nullptr, 0, 0, ys, 0, M, M, 1.0f, 0.0f, 0.0f);
      gemm_wmma_kernel<false, true><<<dim3(M/64, M/64, 1), 128, 0, stream>>>(
          xc, 0, M, ys, 0, M, xc, 0, M, xn, 0, M, M, -1.0f, 2.0f, 0.0f);
      float* t = xc; xc = xn; xn = t;
    }
    // NS_ITERS even -> result in x0s
  }

  // 6) projT = MMT_inv @ Mm   [M,C]
  gemm_wmma_kernel<false, false><<<dim3(C/64, M/64, 1), 128, 0, stream>>>(
      x0s, 0, M, mem_mu, 0, C, nullptr, 0, 0, projT, 0, C, M, 1.0f, 0.0f, 0.0f);

  // 7) w_write = Z[8192,384] @ projT^T  -> [8192,512]
  gemm_wmma_kernel<false, true><<<dim3(M/64, (E*B)/64, 1), 128, 0, stream>>>(
      z, 0, C, projT, 0, C, nullptr, 0, 0, wwr, 0, M, C, 1.0f, 0.0f, 0.0f);

  // 8) G_b = w_b @ w_b^T  (64x64, K=512)
  gemm_wmma_kernel<false, true><<<dim3(1, 1, B), 128, 0, stream>>>(
      wwr, M, B * M, wwr, M, B * M, nullptr, 0, 0,
      G, 64LL * 64, 64, M, 1.0f, 0.0f, 0.0f);

  // 9) Ginv_b
  ginv64_kernel<<<B, 128, 0, stream>>>(G, Ginv);

  // 10) V_b = Ginv_b @ z_b   [64,C]
  gemm_wmma_kernel<false, false><<<dim3(C/64, 1, B), 128, 0, stream>>>(
      Ginv, 64LL * 64, 64, z, C, B * C, nullptr, 0, 0,
      V, 64LL * C, C, 64, 1.0f, 0.0f, 0.0f);

  // 11) new_mem_b = w_b^T @ V_b  [M,C]  -> directly into d_out
  gemm_wmma_kernel<true, false><<<dim3(C/64, M/64, B), 128, 0, stream>>>(
      wwr, M, B * M, V, 64LL * C, C, nullptr, 0, 0,
      newmem_out, (long long)M * C, C, 64, 1.0f, 0.0f, 0.0f);

  // ---------------- Phase 2: read path (reuse scratch) ----------------
  off = 0;
  float* mmt2 = alloc((long long)B * M * M);
  float* xb0  = alloc((long long)B * M * M);
  float* xb1  = alloc((long long)B * M * M);
  float* yb   = alloc((long long)B * M * M);
  float* zmt  = alloc(EBM);
  float* scB  = alloc(B);

  // 12) MMT2_b = nm_b @ nm_b^T + eps*I  (batched, NT)
  gemm_wmma_kernel<false, true><<<dim3(M/64, M/64, B), 128, 0, stream>>>(
      newmem_out, (long long)M * C, C, newmem_out, (long long)M * C, C,
      nullptr, 0, 0, mmt2, (long long)M * M, M, C, 1.0f, 0.0f, REG_EPS);

  // 13-15) batched Newton-Schulz inverse of MMT2 (symmetric -> NT)
  infnorm_scale_kernel<<<B, 256, 0, stream>>>(mmt2, (long long)M * M, M, scB);
  scale_batched_kernel<<<(unsigned)cdiv((long long)B*M*M, 256), 256, 0, stream>>>(
      mmt2, xb0, scB, (long long)M * M, (long long)B * M * M);
  {
    float* xc = xb0; float* xn = xb1;
    for (int it = 0; it < NS_ITERS; ++it) {
      gemm_wmma_kernel<false, true><<<dim3(M/64, M/64, B), 128, 0, stream>>>(
          xc, (long long)M * M, M, mmt2, (long long)M * M, M, nullptr, 0, 0,
          yb, (long long)M * M, M, M, 1.0f, 0.0f, 0.0f);
      gemm_wmma_kernel<false, true><<<dim3(M/64, M/64, B), 128, 0, stream>>>(
          xc, (long long)M * M, M, yb, (long long)M * M, M,
          xc, (long long)M * M, M, xn, (long long)M * M, M, M, -1.0f, 2.0f, 0.0f);
      float* t = xc; xc = xn; xn = t;
    }
    // result in xb0
  }

  // 16) zMt_b = q_b[64,C] @ nm_b^T -> [64,M]
  gemm_wmma_kernel<false, true><<<dim3(M/64, 1, B), 128, 0, stream>>>(
      q_enc, C, B * C, newmem_out, (long long)M * C, C, nullptr, 0, 0,
      zmt, 64LL * M, M, C, 1.0f, 0.0f, 0.0f);

  // 17) w_read = zMt_b @ MMT2_inv_b (symmetric -> NT) -> strided [E,B,M]
  gemm_wmma_kernel<false, true><<<dim3(M/64, 1, B), 128, 0, stream>>>(
      zmt, 64LL * M, M, xb0, (long long)M * M, M, nullptr, 0, 0,
      wread_out, M, B * M, M, 1.0f, 0.0f, 0.0f);

  // 18) w_read += read_noise * exp(0.5*w_logvar)
  add_noise_kernel<<<(unsigned)cdiv(EBM, 256), 256, 0, stream>>>(
      wread_out, r_noise, w_logvar, EBM);

  // 19) retrieved_z = w_read_b[64,M] @ nm_b[M,C] -> strided [E,B,C]
  gemm_wmma_kernel<false, false><<<dim3(C/64, 1, B), 128, 0, stream>>>(
      wread_out, M, B * M, newmem_out, (long long)M * C, C, nullptr, 0, 0,
      retr_out, C, B * C, M, 1.0f, 0.0f, 0.0f);
}